// GCNModel_52682068853201
// MI455X (gfx1250) — compile-verified
//
#include <hip/hip_runtime.h>

// GCN on MI455X (gfx1250, wave32).
// Memory-bound by edge gather/scatter (~1.6GB/layer at 23.3 TB/s); GEMMs are
// tiny (16.4 GFLOP total) so we keep full f32 precision and use the CDNA5
// V_WMMA_F32_16X16X4_F32 matrix pipe for them.

typedef __attribute__((ext_vector_type(2))) float v2f;
typedef __attribute__((ext_vector_type(8))) float v8f;

constexpr int H     = 256;  // hidden width (all conv outputs)
constexpr int FIN   = 128;  // input feature width
constexpr int G     = 64;   // number of graphs
constexpr int A_DIM = 8;    // graph attr width

__device__ __forceinline__ float waveSum(float v) {
  v += __shfl_xor(v, 16, 32);
  v += __shfl_xor(v, 8, 32);
  v += __shfl_xor(v, 4, 32);
  v += __shfl_xor(v, 2, 32);
  v += __shfl_xor(v, 1, 32);
  return v;
}

// ---------------- degree / dinv ----------------
__global__ void k_init_deg(float* deg, int n) {
  int i = blockIdx.x * blockDim.x + threadIdx.x;
  if (i < n) deg[i] = 1.0f;  // self loop
}

__global__ void k_deg_edges(const int* __restrict__ ei, float* deg, int nE) {
  int i = blockIdx.x * blockDim.x + threadIdx.x;
  if (i < nE) unsafeAtomicAdd(&deg[ei[nE + i]], 1.0f);  // dst row
}

__global__ void k_finalize_dinv(float* dinv, int n) {
  int i = blockIdx.x * blockDim.x + threadIdx.x;
  if (i < n) dinv[i] = rsqrtf(dinv[i]);  // deg >= 1 always (self loop)
}

// ---------------- WMMA f32 GEMM: C[M,H] = A[M,K] @ B[K,H] ----------------
// One wave computes a 16x16 tile. A-frag layout (ISA 7.12.2, 32-bit A 16x4):
//   lane = M + 16*(K/2), vgpr = K%2  ->  lane L holds K = {2*(L/16), 2*(L/16)+1}
// B (4x16) mirrors row-striped-across-lanes: lane = N + 16*(K/2), vgpr = K%2.
// C/D: vgpr r -> (M = r + 8*(lane/16), N = lane%16).
__global__ void __launch_bounds__(256)
k_gemm_wmma(const float* __restrict__ A, const float* __restrict__ B,
            float* __restrict__ C, int M, int K) {
  const int tilesN = H / 16;  // 16
  int wave = blockIdx.x * (blockDim.x >> 5) + (threadIdx.x >> 5);
  int tm = wave / tilesN;
  int tn = wave - tm * tilesN;
  if (tm * 16 >= M) return;            // wave-uniform exit: EXEC stays full
  int lane = threadIdx.x & 31;
  int half = lane >> 4;                // 0: K even group low, 1: high
  int mr   = lane & 15;
  int row  = tm * 16 + mr;
  if (row >= M) row = M - 1;           // clamp loads, keep EXEC all-ones
  const float* Ap = A + (size_t)row * K + 2 * half;
  const float* Bp = B + (size_t)(2 * half) * H + tn * 16 + mr;
  v8f c = {};
  for (int k0 = 0; k0 < K; k0 += 4) {
    float2 av = *(const float2*)Ap;    // K = 2*half, 2*half+1
    v2f a, b;
    a.x = av.x;  a.y = av.y;
    b.x = Bp[0]; b.y = Bp[H];
    c = __builtin_amdgcn_wmma_f32_16x16x4_f32(
        /*neg_a=*/false, a, /*neg_b=*/false, b,
        /*c_mod=*/(short)0, c, /*reuse_a=*/false, /*reuse_b=*/false);
    Ap += 4;
    Bp += 4 * H;
  }
  int crow0 = tm * 16 + 8 * half;
  float* Cp = C + (size_t)crow0 * H + tn * 16 + mr;
#pragma unroll
  for (int r = 0; r < 8; ++r)
    if (crow0 + r < M) Cp[(size_t)r * H] = c[r];
}

// ---------------- self-loop init: agg = xw * dinv^2 ----------------
__global__ void k_selfloop(const float* __restrict__ xw,
                           const float* __restrict__ dinv,
                           float* __restrict__ agg, int n) {
  int idx = blockIdx.x * blockDim.x + threadIdx.x;
  if (idx < n) {
    int i = idx >> 8;  // / H
    float d = dinv[i];
    agg[idx] = xw[idx] * d * d;
  }
}

// ---------------- edge scatter: agg[dst] += xw[src] * dinv[src]*dinv[dst] --
__global__ void __launch_bounds__(256)
k_scatter(const int* __restrict__ ei, const float* __restrict__ xw,
          const float* __restrict__ dinv, float* __restrict__ agg, int nE) {
  int e = blockIdx.x * 8 + (threadIdx.x >> 5);  // one wave per edge
  if (e >= nE) return;
  int src = ei[e];
  int dst = ei[nE + e];
  float norm = dinv[src] * dinv[dst];
  int lane = threadIdx.x & 31;
  const float* xr = xw + (size_t)src * H;
  float* ar = agg + (size_t)dst * H;
#pragma unroll
  for (int r = 0; r < 8; ++r) {
    int j = lane + 32 * r;
    unsafeAtomicAdd(&ar[j], xr[j] * norm);
  }
}

// ---------------- bias + LayerNorm + ReLU (wave per node) ----------------
__global__ void __launch_bounds__(256)
k_bias_ln_relu(const float* __restrict__ agg, const float* __restrict__ bias,
               const float* __restrict__ gam, const float* __restrict__ bet,
               float* __restrict__ out, int M) {
  int n = blockIdx.x * 8 + (threadIdx.x >> 5);
  if (n >= M) return;
  int lane = threadIdx.x & 31;
  const float* rowp = agg + (size_t)n * H;
  float v[8];
  float s = 0.f;
#pragma unroll
  for (int r = 0; r < 8; ++r) {
    int j = lane + 32 * r;
    v[r] = rowp[j] + bias[j];
    s += v[r];
  }
  s = waveSum(s);
  float mean = s * (1.0f / H);
  float q = 0.f;
#pragma unroll
  for (int r = 0; r < 8; ++r) { float d = v[r] - mean; q += d * d; }
  q = waveSum(q);
  float inv = rsqrtf(q * (1.0f / H) + 1e-5f);
  float* op = out + (size_t)n * H;
#pragma unroll
  for (int r = 0; r < 8; ++r) {
    int j = lane + 32 * r;
    float y = (v[r] - mean) * inv * gam[j] + bet[j];
    op[j] = y > 0.f ? y : 0.f;
  }
}

// ---------------- pooling + FC head ----------------
__global__ void k_zero(float* p, int n) {
  int i = blockIdx.x * blockDim.x + threadIdx.x;
  if (i < n) p[i] = 0.f;
}

__global__ void __launch_bounds__(256)
k_pool(const float* __restrict__ h, const int* __restrict__ batch,
       float* __restrict__ pooled, float* __restrict__ cnt, int M) {
  int n = blockIdx.x * 8 + (threadIdx.x >> 5);
  if (n >= M) return;
  int lane = threadIdx.x & 31;
  int g = batch[n];
  const float* rowp = h + (size_t)n * H;
  float* pp = pooled + (size_t)g * H;
#pragma unroll
  for (int r = 0; r < 8; ++r) {
    int j = lane + 32 * r;
    unsafeAtomicAdd(&pp[j], rowp[j]);
  }
  if (lane == 0) unsafeAtomicAdd(&cnt[g], 1.0f);
}

__global__ void __launch_bounds__(256)
k_fc1(const float* __restrict__ pooled, const float* __restrict__ cnt,
      const float* __restrict__ attr, const float* __restrict__ W,
      const float* __restrict__ b, float* __restrict__ z) {
  int g = blockIdx.x;   // 64
  int o = threadIdx.x;  // 256
  float c = cnt[g];
  float invc = 1.0f / (c > 1.0f ? c : 1.0f);
  float s = b[o];
  const float* pg = pooled + (size_t)g * H;
  for (int j = 0; j < H; ++j) s += pg[j] * invc * W[j * H + o];
  const float* ag = attr + g * A_DIM;
#pragma unroll
  for (int a = 0; a < A_DIM; ++a) s += ag[a] * W[(H + a) * H + o];
  z[(size_t)g * H + o] = s > 0.f ? s : 0.f;
}

__global__ void k_fc2(const float* __restrict__ z, const float* __restrict__ w2,
                      const float* __restrict__ b2, float* __restrict__ out) {
  int g = threadIdx.x;
  if (g >= G) return;
  float s = b2[0];
  const float* zg = z + (size_t)g * H;
  for (int j = 0; j < H; ++j) s += zg[j] * w2[j];
  out[g] = s;
}

extern "C" void kernel_launch(void* const* d_in, const int* in_sizes, int n_in,
                              void* d_out, int out_size, void* d_ws, size_t ws_size,
                              hipStream_t stream) {
  const float* x    = (const float*)d_in[0];
  const float* attr = (const float*)d_in[1];
  const int*   ei   = (const int*)d_in[2];
  const int*   batch= (const int*)d_in[3];
  const float* W1   = (const float*)d_in[4];
  const float* b1   = (const float*)d_in[5];
  const float* W2   = (const float*)d_in[6];
  const float* b2   = (const float*)d_in[7];
  const float* W3   = (const float*)d_in[8];
  const float* b3   = (const float*)d_in[9];
  const float* g1   = (const float*)d_in[10];
  const float* be1  = (const float*)d_in[11];
  const float* g2   = (const float*)d_in[12];
  const float* be2  = (const float*)d_in[13];
  const float* g3   = (const float*)d_in[14];
  const float* be3  = (const float*)d_in[15];
  const float* fcW1 = (const float*)d_in[16];
  const float* fcb1 = (const float*)d_in[17];
  const float* fcW2 = (const float*)d_in[18];
  const float* fcb2 = (const float*)d_in[19];

  const int N = in_sizes[3];        // 50000 (batch vector length)
  const int E = in_sizes[2] / 2;    // 800000
  const size_t NH = (size_t)N * H;

  float* ws     = (float*)d_ws;
  float* hbuf   = ws;               // [N,H] activations
  float* xw     = ws + NH;          // [N,H] GEMM output
  float* agg    = ws + 2 * NH;      // [N,H] conv accumulation
  float* dinv   = ws + 3 * NH;      // [N]
  float* pooled = dinv + N;         // [G,H]
  float* cnt    = pooled + (size_t)G * H;  // [G]
  float* z      = cnt + G;          // [G,H]
  float* out    = (float*)d_out;    // [G]

  const int TB = 256;

  // degree -> dinv (shared by all 3 layers)
  k_init_deg<<<(N + TB - 1) / TB, TB, 0, stream>>>(dinv, N);
  k_deg_edges<<<(E + TB - 1) / TB, TB, 0, stream>>>(ei, dinv, E);
  k_finalize_dinv<<<(N + TB - 1) / TB, TB, 0, stream>>>(dinv, N);

  auto layer = [&](const float* in, int K, const float* W, const float* b,
                   const float* gam, const float* bet) {
    int tilesM = (N + 15) / 16;
    int waves = tilesM * (H / 16);
    k_gemm_wmma<<<(waves + 7) / 8, TB, 0, stream>>>(in, W, xw, N, K);
    int nh = (int)NH;
    k_selfloop<<<(nh + TB - 1) / TB, TB, 0, stream>>>(xw, dinv, agg, nh);
    k_scatter<<<(E + 7) / 8, TB, 0, stream>>>(ei, xw, dinv, agg, E);
    k_bias_ln_relu<<<(N + 7) / 8, TB, 0, stream>>>(agg, b, gam, bet, hbuf, N);
  };

  layer(x,    FIN, W1, b1, g1, be1);
  layer(hbuf, H,   W2, b2, g2, be2);
  layer(hbuf, H,   W3, b3, g3, be3);

  // mean pool + FC head
  k_zero<<<(G * H + G + TB - 1) / TB, TB, 0, stream>>>(pooled, G * H + G);
  k_pool<<<(N + 7) / 8, TB, 0, stream>>>(hbuf, batch, pooled, cnt, N);
  k_fc1<<<G, H, 0, stream>>>(pooled, cnt, attr, fcW1, fcb1, z);
  k_fc2<<<1, 64, 0, stream>>>(z, fcW2, fcb2, out);
}